// NeighborSearch_batch_4329327034822
// MI455X (gfx1250) — compile-verified
//
#include <hip/hip_runtime.h>

typedef float v2f __attribute__((ext_vector_type(2)));
typedef float v8f __attribute__((ext_vector_type(8)));

#define BATCH   8
#define NDATA   9225
#define MQ      4096
#define MAXN    64
#define NFULL   (NDATA / 16)          // 576 full tiles (no bounds check)
#define QTILES  (MQ / 16)             // 256 query tiles per batch

// Compute one 16x16 squared-distance tile with V_WMMA_F32_16X16X4_F32 and
// compact its hits (ascending index order) into the output lists.
//   A row (query)  = [-2qx, -2qy, |q|^2, 1]
//   B col (data)   = [ dx ,  dy ,   1  , |d|^2]
// C layout: VGPR v -> row v (lanes 0-15) / row v+8 (lanes 16-31), col N = l15.
__device__ __forceinline__ void process_tile(v2f d2, v2f a, bool upper, float r2,
                                             int* cnt, int nidx, unsigned halfsel,
                                             int* __restrict__ baseh)
{
    const float dn = d2.x * d2.x + d2.y * d2.y;
    v2f bm;
    bm.x = upper ? 1.0f : d2.x;         // K0 / K2
    bm.y = upper ? dn   : d2.y;         // K1 / K3

    v8f c = {0.f, 0.f, 0.f, 0.f, 0.f, 0.f, 0.f, 0.f};
    c = __builtin_amdgcn_wmma_f32_16x16x4_f32(
            false, a, false, bm, (short)0, c, false, false);

    #pragma unroll
    for (int v = 0; v < 8; ++v) {
        const bool hit = (c[v] <= r2);
        const unsigned mask = __builtin_amdgcn_ballot_w32(hit);
        const unsigned hmf  = mask & halfsel;           // this half's row only
        // bits strictly below this lane in hmf == within-half prefix count
        const int pos = __builtin_amdgcn_mbcnt_lo(hmf, 0u);
        if (hit) {
            const int s = cnt[v] + pos;
            if (s < MAXN)
                baseh[v * MAXN + s] = nidx;             // v*64 folds into offset
        }
        cnt[v] += __builtin_popcount(hmf);              // uncapped
    }
}

// One wave handles one 16-query tile of one batch, sweeping all data tiles.
__global__ __launch_bounds__(256)
void nbr_search_wmma_kernel(const float* __restrict__ data,
                            const float* __restrict__ queries,
                            const float* __restrict__ radius,
                            int* __restrict__ nb_index,
                            int* __restrict__ counts)
{
    const int lane      = threadIdx.x & 31;
    const int waveInBlk = threadIdx.x >> 5;
    const int w         = blockIdx.x * 8 + waveInBlk;
    const int b         = w / QTILES;
    const int qtile     = w % QTILES;
    if (b >= BATCH) return;                     // wave-uniform guard

    const int l15  = lane & 15;                 // M (A) / N (B,C) position
    const int half = lane >> 4;                 // 0: lanes 0-15, 1: lanes 16-31
    const bool upper = (half != 0);
    const unsigned halfsel = upper ? 0xFFFF0000u : 0x0000FFFFu;

    const float r  = radius[0];
    const float r2 = r * r;

    // Base of this tile's 16 output rows; baseh points at this half's 8 rows.
    int* base  = nb_index + ((size_t)(b * MQ + qtile * 16)) * MAXN;
    int* baseh = base + half * 8 * MAXN;

    // Pre-fill 16*64 slots with -1 using b128 stores (same-wave stores stay
    // ordered with the later hit-stores).
    {
        int4* base4 = (int4*)base;
        const int4 m1 = make_int4(-1, -1, -1, -1);
        #pragma unroll
        for (int i = 0; i < 8; ++i) base4[i * 32 + lane] = m1;
    }

    // ---- A operand: 16x4 F32 layout ----
    // lanes 0-15:  VGPR0=K0, VGPR1=K1 ; lanes 16-31: VGPR0=K2, VGPR1=K3
    const float* qp = queries + ((size_t)b * MQ + qtile * 16 + l15) * 2;
    const float qx = qp[0], qy = qp[1];
    const float qn = qx * qx + qy * qy;
    v2f a;
    a.x = upper ? qn   : (-2.0f * qx);
    a.y = upper ? 1.0f : (-2.0f * qy);

    int cnt[8];
    #pragma unroll
    for (int v = 0; v < 8; ++v) cnt[v] = 0;

    // ---- 576 full tiles, software-pipelined: load t+1 before processing t
    // so the s_wait_loadcnt lands after a full tile of compaction work. ----
    const float* dptr = data + ((size_t)b * NDATA + l15) * 2;
    v2f d2 = *(const v2f*)dptr;
    for (int t = 0; t < NFULL - 1; ++t) {
        const v2f d2n = *(const v2f*)(dptr + (size_t)(t + 1) * 32);
        process_tile(d2, a, upper, r2, cnt, t * 16 + l15, halfsel, baseh);
        d2 = d2n;
    }
    process_tile(d2, a, upper, r2, cnt, (NFULL - 1) * 16 + l15, halfsel, baseh);

    // ---- Epilogue tile (9 valid points), guarded load ----
    {
        const int nidx = NFULL * 16 + l15;
        v2f d2e;
        d2e.x = 1.0e9f;                 // sentinel: never within radius
        d2e.y = 1.0e9f;
        if (nidx < NDATA) {
            const float* dp = data + ((size_t)b * NDATA + nidx) * 2;
            d2e.x = dp[0];
            d2e.y = dp[1];
        }
        process_tile(d2e, a, upper, r2, cnt, nidx, halfsel, baseh);
    }

    // One lane per half writes the 8 row counts it owns.
    if (l15 == 0) {
        #pragma unroll
        for (int v = 0; v < 8; ++v)
            counts[b * MQ + qtile * 16 + v + 8 * half] = cnt[v];
    }
}

// Per-batch exclusive-scan into row_splits [B, m+1]. 4096 serial adds per
// batch: negligible next to the main kernel.
__global__ void row_splits_kernel(const int* __restrict__ counts,
                                  int* __restrict__ rs)
{
    const int b = blockIdx.x * blockDim.x + threadIdx.x;
    if (b >= BATCH) return;
    int* r = rs + (size_t)b * (MQ + 1);
    int acc = 0;
    r[0] = 0;
    for (int q = 0; q < MQ; ++q) {
        acc += counts[b * MQ + q];
        r[q + 1] = acc;
    }
}

extern "C" void kernel_launch(void* const* d_in, const int* in_sizes, int n_in,
                              void* d_out, int out_size, void* d_ws, size_t ws_size,
                              hipStream_t stream) {
    const float* data    = (const float*)d_in[0];   // [B, n, 2]
    const float* queries = (const float*)d_in[1];   // [B, m, 2]
    const float* radius  = (const float*)d_in[2];   // [1]

    int* nb_index = (int*)d_out;                                  // [B, m, 64]
    int* rs       = nb_index + (size_t)BATCH * MQ * MAXN;         // [B, m+1]
    int* counts   = (int*)d_ws;                                   // [B, m] scratch

    // 2048 waves total, 8 waves (256 threads) per block -> 256 blocks.
    const int nWaves = BATCH * QTILES;
    const int blocks = (nWaves + 7) / 8;
    nbr_search_wmma_kernel<<<blocks, 256, 0, stream>>>(data, queries, radius,
                                                       nb_index, counts);
    row_splits_kernel<<<1, BATCH, 0, stream>>>(counts, rs);
}